// RefineMultiBoxLoss_77893526880381
// MI455X (gfx1250) — compile-verified
//
#include <hip/hip_runtime.h>
#include <math.h>

typedef __attribute__((ext_vector_type(2))) float v2f;
typedef __attribute__((ext_vector_type(8))) float v8f;

#define NUMC   81
#define PADC   84
#define THRESH 0.5f
#define NEGPOS 3

__device__ __forceinline__ float sl1(float x) {
    float ax = fabsf(x);
    return (ax < 1.0f) ? 0.5f * ax * ax : ax - 0.5f;
}

template <int BS>
__device__ __forceinline__ float blockReduceSum(float v, float* red) {
    int t = threadIdx.x;
    red[t] = v;
    __syncthreads();
#pragma unroll
    for (int s = BS / 2; s > 0; s >>= 1) {
        if (t < s) red[t] += red[t + s];
        __syncthreads();
    }
    float r = red[0];
    __syncthreads();
    return r;
}

__device__ __forceinline__ float pick8(const v8f& a, int i) {
    float s = a[0];
    s = (i == 1) ? a[1] : s;
    s = (i == 2) ? a[2] : s;
    s = (i == 3) ? a[3] : s;
    s = (i == 4) ? a[4] : s;
    s = (i == 5) ? a[5] : s;
    s = (i == 6) ? a[6] : s;
    s = (i == 7) ? a[7] : s;
    return s;
}

// ---------------------------------------------------------------- init
__global__ void k_init(unsigned long long* bestPrior, int* num_pos, int nBP, int B) {
    int i = blockIdx.x * 256 + threadIdx.x;
    if (i < nBP) bestPrior[i] = 0ull;
    if (i < B) num_pos[i] = 0;
}

// ---------------------------------------------------------------- match (IoU, best per prior, best per truth)
__global__ void k_match(const float* __restrict__ priors, const float* __restrict__ truths,
                        float* __restrict__ bto, int* __restrict__ bti,
                        unsigned long long* __restrict__ bestPrior, int P, int O) {
    int b = blockIdx.y;
    int p = blockIdx.x * 256 + threadIdx.x;
    __shared__ float tx[64 * 4];               // supports O <= 64
    __shared__ unsigned long long key[64];
    int tid = threadIdx.x;
    if (tid < O * 4) tx[tid] = truths[(size_t)b * O * 4 + tid];
    if (tid < O) key[tid] = 0ull;
    __syncthreads();

    if (p < P) {
        float cx = priors[p * 4 + 0], cy = priors[p * 4 + 1];
        float w  = priors[p * 4 + 2], h  = priors[p * 4 + 3];
        float px1 = cx - 0.5f * w, py1 = cy - 0.5f * h;
        float px2 = cx + 0.5f * w, py2 = cy + 0.5f * h;
        float pa = (px2 - px1) * (py2 - py1);
        float best = -1.0f;
        int bestO = 0;
        for (int o = 0; o < O; ++o) {
            float tx1 = tx[o * 4 + 0], ty1 = tx[o * 4 + 1];
            float tx2 = tx[o * 4 + 2], ty2 = tx[o * 4 + 3];
            float lx = fmaxf(px1, tx1), ly = fmaxf(py1, ty1);
            float rx = fminf(px2, tx2), ry = fminf(py2, ty2);
            float iw = fmaxf(rx - lx, 0.0f), ih = fmaxf(ry - ly, 0.0f);
            float inter = iw * ih;
            float ta = (tx2 - tx1) * (ty2 - ty1);
            float iou = inter / (pa + ta - inter);
            // pack (iou bits, ~p) so ties pick lowest prior index (argmax semantics)
            unsigned long long kk =
                (((unsigned long long)__float_as_uint(iou)) << 32) |
                (unsigned)(0xFFFFFFFFu - (unsigned)p);
            atomicMax(&key[o], kk);
            if (iou > best) { best = iou; bestO = o; }
        }
        bto[(size_t)b * P + p] = best;
        bti[(size_t)b * P + p] = bestO;
    }
    __syncthreads();
    if (tid < O) atomicMax(&bestPrior[(size_t)b * O + tid], key[tid]);
}

// ---------------------------------------------------------------- force-match overrides (sequential: last truth wins)
__global__ void k_override(const unsigned long long* __restrict__ bestPrior,
                           float* __restrict__ bto, int* __restrict__ bti, int P, int O) {
    int b = blockIdx.x;
    if (threadIdx.x == 0) {
        for (int o = 0; o < O; ++o) {
            unsigned long long v = bestPrior[(size_t)b * O + o];
            unsigned p = 0xFFFFFFFFu - (unsigned)(v & 0xFFFFFFFFull);
            bto[(size_t)b * P + p] = 2.0f;
            bti[(size_t)b * P + p] = o;
        }
    }
}

// ---------------------------------------------------------------- conf_t, loc loss, obj loss, mining_obj
__global__ void k_locobj(const float* __restrict__ loc, const float* __restrict__ obj,
                         const float* __restrict__ priors, const float* __restrict__ truths,
                         const int* __restrict__ labels, const float* __restrict__ bto,
                         const int* __restrict__ bti, int* __restrict__ conf_t,
                         float* __restrict__ mining_obj, int* __restrict__ num_pos,
                         float* __restrict__ partials, int P, int O) {
    int b = blockIdx.y;
    int p = blockIdx.x * 256 + threadIdx.x;
    float lossL = 0.0f, objPos = 0.0f;
    if (p < P) {
        size_t bp = (size_t)b * P + p;
        float ov = bto[bp];
        int ti = bti[bp];
        int ct = (ov < THRESH) ? 0 : labels[(size_t)b * O + ti];
        conf_t[bp] = ct;
        bool pos = ct > 0;
        if (pos) atomicAdd(&num_pos[b], 1);
        // objectness CE (2 logits)
        float o0 = obj[bp * 2 + 0], o1 = obj[bp * 2 + 1];
        float m = fmaxf(o0, o1);
        float lse = m + __logf(__expf(o0 - m) + __expf(o1 - m));
        float ce = lse - (pos ? o1 : o0);
        mining_obj[bp] = pos ? 0.0f : ce;
        objPos = pos ? ce : 0.0f;
        if (pos) {
            float cx = priors[p * 4 + 0], cy = priors[p * 4 + 1];
            float pw = priors[p * 4 + 2], ph = priors[p * 4 + 3];
            const float* t = &truths[((size_t)b * O + ti) * 4];
            float mx = 0.5f * (t[0] + t[2]), my = 0.5f * (t[1] + t[3]);
            float mw = t[2] - t[0], mh = t[3] - t[1];
            float g0 = (mx - cx) / (0.1f * pw);
            float g1 = (my - cy) / (0.1f * ph);
            float g2 = __logf(mw / pw) / 0.2f;
            float g3 = __logf(mh / ph) / 0.2f;
            const float* ld = &loc[bp * 4];
            lossL = sl1(ld[0] - g0) + sl1(ld[1] - g1) + sl1(ld[2] - g2) + sl1(ld[3] - g3);
        }
    }
    __shared__ float red[256];
    float tl = blockReduceSum<256>(lossL, red);
    float to = blockReduceSum<256>(objPos, red);
    if (threadIdx.x == 0) {
        int blk = blockIdx.y * gridDim.x + blockIdx.x;
        partials[blk * 2 + 0] = tl;
        partials[blk * 2 + 1] = to;
    }
}

// ---------------------------------------------------------------- ce_c via LDS staging + WMMA f32 row-sum of exp
// block = 128 (4 waves); each wave handles 16 priors (rows) of 81 classes.
__global__ void k_cec(const float* __restrict__ conf, const int* __restrict__ conf_t,
                      float* __restrict__ mining_c, float* __restrict__ partials, int P) {
    int b = blockIdx.y;
    int wave = threadIdx.x >> 5;
    int lane = threadIdx.x & 31;
    int rowBase = blockIdx.x * 64 + wave * 16;

    __shared__ float L[4 * 16 * PADC];
    float* Lw = &L[wave * 16 * PADC];
    const float* rowsrc = conf + ((size_t)b * P) * NUMC;

    // Stage A: coalesced global -> LDS (16 rows), prefetch next group
#pragma unroll
    for (int r = 0; r < 16; ++r) {
        int pr = rowBase + r;
        if (pr >= P) pr = P - 1;
        const float* src = rowsrc + (size_t)pr * NUMC;
        __builtin_prefetch(src + (size_t)16 * NUMC, 0, 3);  // global_prefetch_b8
#pragma unroll
        for (int j = 0; j < 3; ++j) {
            int c = lane + 32 * j;
            if (c < NUMC) Lw[r * PADC + c] = src[c];
        }
    }
    __syncthreads();

    int r = lane & 15;       // row owned for max / A-matrix
    int hlf = lane >> 4;     // K-half per ISA A layout (lanes 0-15: K0,K1; 16-31: K2,K3)

    // Stage B: per-row max (two lanes split the row, combine across halves)
    float mx = -3.402823466e38f;
    int cbeg = hlf ? 41 : 0;
    int cend = hlf ? NUMC : 41;
    for (int c = cbeg; c < cend; ++c) mx = fmaxf(mx, Lw[r * PADC + c]);
    mx = fmaxf(mx, __shfl_xor(mx, 16));

    // Stage C: D += A(16x4 of exp) x B(4x16 ones)  => every D column = row sum
    v8f acc = {0.f, 0.f, 0.f, 0.f, 0.f, 0.f, 0.f, 0.f};
    v2f bones;
    bones.x = 1.0f; bones.y = 1.0f;
#pragma unroll
    for (int j = 0; j < 21; ++j) {
        int c0 = 4 * j + 2 * hlf;
        int c0c = (c0 < NUMC) ? c0 : (NUMC - 1);
        int c1c = (c0 + 1 < NUMC) ? c0 + 1 : (NUMC - 1);
        float a0 = __expf(Lw[r * PADC + c0c] - mx);
        float a1 = __expf(Lw[r * PADC + c1c] - mx);
        if (c0 >= NUMC) a0 = 0.0f;
        if (c0 + 1 >= NUMC) a1 = 0.0f;
        v2f A;
        A.x = a0; A.y = a1;
        acc = __builtin_amdgcn_wmma_f32_16x16x4_f32(false, A, false, bones,
                                                    (short)0, acc, false, false);
    }

    // Finalize: D layout -> lanes 0-15 hold rows 0-7 sums in v[0..7], lanes 16-31 rows 8-15.
    bool fin = (lane < 8) || (lane >= 16 && lane < 24);
    int idx = lane & 7;
    int row = (lane < 16) ? idx : idx + 8;
    float s = pick8(acc, idx);
    float mrow = __shfl(mx, row & 15);  // lane (row&15) owns max of row (lane&15)

    float contrib = 0.0f;
    int p = rowBase + row;
    if (fin && p < P) {
        size_t bp = (size_t)b * P + p;
        int tgt = conf_t[bp];
        float picked = Lw[row * PADC + tgt];
        float ce = mrow + __logf(s) - picked;
        bool pos = tgt > 0;
        mining_c[bp] = pos ? 0.0f : ce;
        contrib = pos ? ce : 0.0f;
    }

    __shared__ float red[128];
    float tc = blockReduceSum<128>(contrib, red);
    if (threadIdx.x == 0) partials[blockIdx.y * gridDim.x + blockIdx.x] = tc;
}

// ---------------------------------------------------------------- exact radix-select top-k sum (deterministic fixed-point)
__global__ void k_topk(const float* __restrict__ mining_obj, const float* __restrict__ mining_c,
                       const int* __restrict__ num_pos, float* __restrict__ topk, int P) {
    int b = blockIdx.x >> 1;
    int which = blockIdx.x & 1;  // 0 = obj, 1 = conf
    const float* v = which ? (mining_c + (size_t)b * P) : (mining_obj + (size_t)b * P);
    long long kk = (long long)num_pos[b] * NEGPOS;
    if (kk > P - 1) kk = P - 1;
    int k = (int)kk;

    __shared__ unsigned long long sfix[256];
    __shared__ unsigned cnt[256];
    __shared__ unsigned prefix_s, mask_s;
    __shared__ int want_s;
    __shared__ unsigned long long tot_s;
    if (threadIdx.x == 0) { prefix_s = 0u; mask_s = 0u; want_s = k; tot_s = 0ull; }
    __syncthreads();
    if (k <= 0) {
        if (threadIdx.x == 0) topk[blockIdx.x] = 0.0f;
        return;
    }
    for (int pass = 3; pass >= 0; --pass) {
        cnt[threadIdx.x] = 0u;
        sfix[threadIdx.x] = 0ull;
        __syncthreads();
        unsigned pref = prefix_s, msk = mask_s;
        for (int i = threadIdx.x; i < P; i += 256) {
            float x = v[i];                          // all values >= 0
            unsigned bits = __float_as_uint(x);
            if ((bits & msk) == pref) {
                unsigned bin = (bits >> (8 * pass)) & 255u;
                atomicAdd(&cnt[bin], 1u);
                atomicAdd(&sfix[bin], (unsigned long long)((double)x * 4294967296.0));
            }
        }
        __syncthreads();
        if (threadIdx.x == 0) {
            int want = want_s;
            unsigned long long tot = tot_s;
            int chosen = 0;
            unsigned c_hi = 0;
            for (int bin = 255; bin >= 0; --bin) {
                if (c_hi + cnt[bin] >= (unsigned)want) { chosen = bin; break; }
                c_hi += cnt[bin];
                tot += sfix[bin];
            }
            want_s = want - (int)c_hi;
            tot_s = tot;
            prefix_s = pref | ((unsigned)chosen << (8 * pass));
            mask_s = msk | (255u << (8 * pass));
        }
        __syncthreads();
    }
    if (threadIdx.x == 0) {
        float val = __uint_as_float(prefix_s);  // exact k-th value; ties included at this value
        unsigned long long tot =
            tot_s + (unsigned long long)want_s * (unsigned long long)((double)val * 4294967296.0);
        topk[blockIdx.x] = (float)((double)tot / 4294967296.0);
    }
}

// ---------------------------------------------------------------- final combine
__global__ void k_final(const float* __restrict__ partials3, int n3,
                        const float* __restrict__ partials4, int n4,
                        const int* __restrict__ num_pos, const float* __restrict__ topk,
                        float* __restrict__ out, int B, int P) {
    __shared__ float red[256];
    float a = 0.0f, bb = 0.0f, c = 0.0f;
    for (int i = threadIdx.x; i < n3; i += 256) {
        a += partials3[i * 2 + 0];
        bb += partials3[i * 2 + 1];
    }
    for (int i = threadIdx.x; i < n4; i += 256) c += partials4[i];
    a = blockReduceSum<256>(a, red);
    bb = blockReduceSum<256>(bb, red);
    c = blockReduceSum<256>(c, red);
    if (threadIdx.x == 0) {
        long long sumNp = 0, N1 = 0;
        float topObj = 0.0f, topC = 0.0f;
        for (int i = 0; i < B; ++i) {
            int np = num_pos[i];
            sumNp += np;
            long long nn = (long long)np * NEGPOS;
            if (nn > P - 1) nn = P - 1;
            N1 += nn;
            topObj += topk[i * 2 + 0];
            topC += topk[i * 2 + 1];
        }
        float N = (float)(sumNp > 0 ? sumNp : 1);
        float N1f = (float)(N1 > 0 ? N1 : 1);
        out[0] = a / N;
        out[1] = (c + topC) / N;
        out[2] = 0.4f * (bb + topObj) / N1f;
    }
}

// ---------------------------------------------------------------- host launch
extern "C" void kernel_launch(void* const* d_in, const int* in_sizes, int n_in,
                              void* d_out, int out_size, void* d_ws, size_t ws_size,
                              hipStream_t stream) {
    const float* loc = (const float*)d_in[0];
    const float* conf = (const float*)d_in[1];
    const float* obj = (const float*)d_in[2];
    const float* priors = (const float*)d_in[3];
    const float* truths = (const float*)d_in[4];
    const int* labels = (const int*)d_in[5];

    int P = in_sizes[3] / 4;
    int B = in_sizes[0] / (P * 4);
    int O = in_sizes[5] / B;
    float* out = (float*)d_out;

    char* w = (char*)d_ws;
    auto alloc = [&](size_t bytes) -> void* {
        void* r = (void*)w;
        w += (bytes + 255) & ~(size_t)255;
        return r;
    };
    size_t BP = (size_t)B * P;
    float* bto = (float*)alloc(BP * 4);
    int* bti = (int*)alloc(BP * 4);
    int* conf_t = (int*)alloc(BP * 4);
    float* mining_obj = (float*)alloc(BP * 4);
    float* mining_c = (float*)alloc(BP * 4);
    unsigned long long* bestPrior = (unsigned long long*)alloc((size_t)B * O * 8);
    int* num_pos = (int*)alloc((size_t)B * 4);
    float* topk = (float*)alloc((size_t)2 * B * 4);
    int pblk = (P + 255) / 256;
    int n3 = pblk * B;
    float* partials3 = (float*)alloc((size_t)n3 * 2 * 4);
    int cblk = (P + 63) / 64;
    int n4 = cblk * B;
    float* partials4 = (float*)alloc((size_t)n4 * 4);

    int nInit = (B * O > B) ? B * O : B;
    k_init<<<(nInit + 255) / 256, 256, 0, stream>>>(bestPrior, num_pos, B * O, B);

    dim3 g1(pblk, B);
    k_match<<<g1, 256, 0, stream>>>(priors, truths, bto, bti, bestPrior, P, O);
    k_override<<<B, 32, 0, stream>>>(bestPrior, bto, bti, P, O);
    k_locobj<<<g1, 256, 0, stream>>>(loc, obj, priors, truths, labels, bto, bti,
                                     conf_t, mining_obj, num_pos, partials3, P, O);
    dim3 g4(cblk, B);
    k_cec<<<g4, 128, 0, stream>>>(conf, conf_t, mining_c, partials4, P);
    k_topk<<<2 * B, 256, 0, stream>>>(mining_obj, mining_c, num_pos, topk, P);
    k_final<<<1, 256, 0, stream>>>(partials3, n3, partials4, n4, num_pos, topk, out, B, P);
}